// AttentionLayer_11562051961156
// MI455X (gfx1250) — compile-verified
//
#include <hip/hip_runtime.h>

typedef __attribute__((ext_vector_type(2))) float v2f;
typedef __attribute__((ext_vector_type(8))) float v8f;

#define B_   8
#define N_   2048   // N_IN == N_OUT
#define K_   64
#define V_   64
#define ITILES (N_ / 16)   // 128 i-tiles
#define KSTEPS (K_ / 4)    // 16 wmma k-steps (K=4 each)

// Compute one 16(i) x 16(o) logit tile via a chained f32 WMMA:
//   D[M=i, N=o] = sum_k key[i,k] * (query[o,k] * 1/sqrt(K))   (scale pre-folded into qb)
// A frag (16x4, 32-bit): lanes0-15 M=lane, V0:K=0 V1:K=1 ; lanes16-31 V0:K=2 V1:K=3
// B frag (4x16, 32-bit): lanes0-15 N=lane, V0:K=0 V1:K=1 ; lanes16-31 V0:K=2 V1:K=3
__device__ __forceinline__ v8f logits_tile(const float* __restrict__ krow,
                                           const v2f* __restrict__ qb) {
    v8f c = {0.f, 0.f, 0.f, 0.f, 0.f, 0.f, 0.f, 0.f};
#pragma unroll
    for (int kk = 0; kk < KSTEPS; ++kk) {
        v2f a;
        a.x = krow[4 * kk + 0];
        a.y = krow[4 * kk + 1];
        c = __builtin_amdgcn_wmma_f32_16x16x4_f32(false, a, false, qb[kk],
                                                  (short)0, c, false, false);
    }
    return c;
}

__launch_bounds__(32)
__global__ void attn_fused(const float* __restrict__ key,     // [B, N, 1, K]
                           const float* __restrict__ query,   // [B, 1, N, K]
                           const float* __restrict__ value,   // [B, N, V]
                           float* __restrict__ out_values,    // [B, V, N]
                           float* __restrict__ out_weights)   // [B, N, N]
{
    __shared__ float lds_w[16 * 16];   // W tile staging (i-major) for transpose
    __shared__ float lds_o[64 * 16];   // out tile, v-major for conflict-free reads
    __shared__ float lds_n[16];        // per-o inverse L2 norm

    const int wg    = blockIdx.x;           // 0 .. B*(N/16)-1
    const int b     = wg / (N_ / 16);
    const int ot    = wg % (N_ / 16);
    const int obase = ot * 16;

    const int lane = threadIdx.x;           // 0..31 (wave32)
    const int half = lane >> 4;             // 0 | 1
    const int l16  = lane & 15;

    const float* keyB = key   + (size_t)b * N_ * K_;
    const float* qryB = query + (size_t)b * N_ * K_;
    const float* valB = value + (size_t)b * N_ * V_;
    float* wB = out_weights + (size_t)b * N_ * N_;
    float* oB = out_values  + (size_t)b * V_ * N_;

    // ---- load query B-fragments once, with 1/sqrt(K) folded in ----
    v2f qb[KSTEPS];
    {
        const float scale = 0.125f; // 1/sqrt(64)
        const float* qrow = qryB + (size_t)(obase + l16) * K_ + 2 * half;
#pragma unroll
        for (int kk = 0; kk < KSTEPS; ++kk) {
            qb[kk].x = qrow[4 * kk + 0] * scale;
            qb[kk].y = qrow[4 * kk + 1] * scale;
        }
    }

    // ---- pass 1: online softmax stats over i for column o = obase+l16 ----
    // each lane tracks its half of the 16 rows of every tile (M=r+8*half)
    float m = -1e30f;
    float s = 0.f;

    for (int it = 0; it < ITILES; ++it) {
        const float* krow = keyB + (size_t)(it * 16 + l16) * K_ + 2 * half;
        v8f c = logits_tile(krow, qb);

        float tmax = c[0];
#pragma unroll
        for (int r = 1; r < 8; ++r) tmax = fmaxf(tmax, c[r]);
        const float mn = fmaxf(m, tmax);
        float acc = 0.f;
#pragma unroll
        for (int r = 0; r < 8; ++r) acc += __expf(c[r] - mn);
        s = s * __expf(m - mn) + acc;
        m = mn;
    }
    // combine the two lane-halves (rows 0-7 vs 8-15 of each tile)
    {
        const float m2 = __shfl_xor(m, 16);
        const float s2 = __shfl_xor(s, 16);
        const float mn = fmaxf(m, m2);
        s = s * __expf(m - mn) + s2 * __expf(m2 - mn);
        m = mn;
    }
    // softmax: w = exp(c - m)/s = exp(c - (m + ln s)) — one fused offset
    const float mls = m + __logf(s);

    // ---- pass 2: recompute logits, emit weights, accumulate out[o,v] ----
    // acc2[nt]: D[M=o(16), N=v(16)] for v-tile nt (V=64 -> 4 tiles)
    v8f acc2[4];
#pragma unroll
    for (int nt = 0; nt < 4; ++nt)
        acc2[nt] = (v8f){0.f, 0.f, 0.f, 0.f, 0.f, 0.f, 0.f, 0.f};

    for (int it = 0; it < ITILES; ++it) {
        const float* krow = keyB + (size_t)(it * 16 + l16) * K_ + 2 * half;
        v8f c = logits_tile(krow, qb);

        float w[8];
#pragma unroll
        for (int r = 0; r < 8; ++r)
            w[r] = __expf(c[r] - mls);

        // coalesced weights store: weights[b, i, obase+l16]
#pragma unroll
        for (int r = 0; r < 8; ++r)
            wB[(size_t)(it * 16 + r + 8 * half) * N_ + obase + l16] = w[r];

        // stage W tile (i-major) in LDS for the transposed A-fragments
#pragma unroll
        for (int r = 0; r < 8; ++r)
            lds_w[(r + 8 * half) * 16 + l16] = w[r];
        __syncthreads();

        // out[o,v] += W^T (16o x 16i) @ value (16i x 64v), f32 WMMA, K=4 chunks
#pragma unroll
        for (int kc = 0; kc < 4; ++kc) {
            v2f aw;  // A[M=o, K=i]: lane -> o=l16, K = 4*kc + 2*half (+1)
            aw.x = lds_w[(4 * kc + 2 * half + 0) * 16 + l16];
            aw.y = lds_w[(4 * kc + 2 * half + 1) * 16 + l16];
#pragma unroll
            for (int nt = 0; nt < 4; ++nt) {
                v2f bv;  // B[K=i, N=v]: lane -> v = nt*16+l16
                const float* vrow = valB +
                    (size_t)(it * 16 + 4 * kc + 2 * half) * V_ + nt * 16 + l16;
                bv.x = vrow[0];
                bv.y = vrow[V_];
                acc2[nt] = __builtin_amdgcn_wmma_f32_16x16x4_f32(
                    false, aw, false, bv, (short)0, acc2[nt], false, false);
            }
        }
        __syncthreads();  // lds_w reused next iteration
    }

    // ---- finalize: L2 normalize over v per o, write values[b, v, o] ----
#pragma unroll
    for (int nt = 0; nt < 4; ++nt)
#pragma unroll
        for (int r = 0; r < 8; ++r)
            lds_o[(nt * 16 + l16) * 16 + (r + 8 * half)] = acc2[nt][r];
    __syncthreads();

    float ss = 0.f;
    for (int v = half * 32; v < half * 32 + 32; ++v) {
        const float x = lds_o[v * 16 + l16];
        ss += x * x;
    }
    ss += __shfl_xor(ss, 16);
    const float inv_n = 1.f / (sqrtf(ss) + 1e-12f);
    if (lane < 16) lds_n[lane] = inv_n;
    __syncthreads();

    for (int t = 0; t < 32; ++t) {
        const int idx = t * 32 + lane;   // 1024 = 16(o) x 64(v)
        const int o = idx & 15;
        const int v = idx >> 4;
        oB[(size_t)v * N_ + obase + o] = lds_o[v * 16 + o] * lds_n[o];
    }
}

extern "C" void kernel_launch(void* const* d_in, const int* in_sizes, int n_in,
                              void* d_out, int out_size, void* d_ws, size_t ws_size,
                              hipStream_t stream) {
    const float* key   = (const float*)d_in[0];
    const float* query = (const float*)d_in[1];
    const float* value = (const float*)d_in[2];

    float* out_values  = (float*)d_out;                       // [B, V, N]
    float* out_weights = out_values + (size_t)B_ * V_ * N_;   // [B, N, N]

    dim3 grid(B_ * (N_ / 16));
    dim3 block(32);
    attn_fused<<<grid, block, 0, stream>>>(key, query, value, out_values, out_weights);
}